// AttentionLayer_85383949844589
// MI455X (gfx1250) — compile-verified
//
#include <hip/hip_runtime.h>
#include <hip/hip_bf16.h>

typedef __attribute__((ext_vector_type(2))) float v2f;
typedef __attribute__((ext_vector_type(4))) float v4f;
typedef __attribute__((ext_vector_type(8))) float v8f;

#define Bsz 16
#define Csz 128
#define Ksz 128
#define Dsz 256

// ---------------------------------------------------------------------------
// Kernel 1: Out[m][n] = sum_d A[m][d] * Wt[n][d]   (A: 2048x256, Wt: 256x256)
// One 16x16 tile per wave via V_WMMA_F32_16X16X4_F32, K-loop of 64 steps.
// ---------------------------------------------------------------------------
__global__ void gemm_f32_wmma(const float* __restrict__ A,
                              const float* __restrict__ Wt,
                              float* __restrict__ Out) {
  int wave = threadIdx.x >> 5;
  int lane = threadIdx.x & 31;
  int tile = blockIdx.x * 4 + wave;      // 2048 tiles: 128 m-tiles x 16 n-tiles
  int m_tile = tile >> 4;
  int n_tile = tile & 15;
  int half = lane >> 4;                  // 0: K=0,1  1: K=2,3
  int l16  = lane & 15;

  const float* arow = A  + (size_t)(m_tile * 16 + l16) * Dsz + 2 * half;
  const float* brow = Wt + (size_t)(n_tile * 16 + l16) * Dsz + 2 * half;

  v8f acc = {};
  for (int d0 = 0; d0 < Dsz; d0 += 4) {
    v2f a = *(const v2f*)(arow + d0);
    v2f b = *(const v2f*)(brow + d0);
    acc = __builtin_amdgcn_wmma_f32_16x16x4_f32(false, a, false, b,
                                                (short)0, acc, false, false);
  }
  int col = n_tile * 16 + l16;
  int row0 = m_tile * 16 + 8 * half;
#pragma unroll
  for (int r = 0; r < 8; ++r)
    Out[(size_t)(row0 + r) * Dsz + col] = acc[r];
}

// ---------------------------------------------------------------------------
// Kernel 2: BN stats per channel (256 channels: 128 for g0, 128 for g1).
// Fold gamma/beta into scale/shift.
// ---------------------------------------------------------------------------
__global__ void bn_stats(const float* __restrict__ G0, const float* __restrict__ G1,
                         const float* __restrict__ g0g, const float* __restrict__ g0b,
                         const float* __restrict__ g1g, const float* __restrict__ g1b,
                         float* __restrict__ scale0, float* __restrict__ shift0,
                         float* __restrict__ scale1, float* __restrict__ shift1) {
  __shared__ float ssum[256];
  __shared__ float ssq[256];
  int ch = blockIdx.x;
  const float *G, *gam, *bet;
  float *sc, *sh;
  int c;
  if (ch < 128) { G = G0; gam = g0g; bet = g0b; sc = scale0; sh = shift0; c = ch; }
  else          { G = G1; gam = g1g; bet = g1b; sc = scale1; sh = shift1; c = ch - 128; }

  int d = threadIdx.x;
  float s = 0.f, q = 0.f;
#pragma unroll
  for (int b = 0; b < Bsz; ++b) {
    float v = G[(size_t)(b * 128 + c) * Dsz + d];
    s += v; q += v * v;
  }
  ssum[d] = s; ssq[d] = q;
  __syncthreads();
  for (int off = 128; off > 0; off >>= 1) {
    if (d < off) { ssum[d] += ssum[d + off]; ssq[d] += ssq[d + off]; }
    __syncthreads();
  }
  if (d == 0) {
    const float inv_n = 1.f / 4096.f;
    float mean = ssum[0] * inv_n;
    float var  = ssq[0] * inv_n - mean * mean;
    float scl  = gam[c] * rsqrtf(var + 1e-5f);
    sc[c] = scl;
    sh[c] = bet[c] - mean * scl;
  }
}

// ---------------------------------------------------------------------------
// Kernel 3: apply BN + sigmoid; build s0, s1, t1m = kc * s1 * kmask.
// ---------------------------------------------------------------------------
__global__ void bn_sig_apply(const float* __restrict__ G0, const float* __restrict__ G1,
                             const float* __restrict__ kc,
                             const float* __restrict__ scale0, const float* __restrict__ shift0,
                             const float* __restrict__ scale1, const float* __restrict__ shift1,
                             const int* __restrict__ kmask,
                             float* __restrict__ s0, float* __restrict__ s1,
                             float* __restrict__ t1m) {
  int i = blockIdx.x * 256 + threadIdx.x;       // [0, 2*524288)
  const int half_n = Bsz * Csz * Dsz;           // 524288
  if (i < half_n) {
    int c = (i >> 8) & 127;
    float g = G0[i] * scale0[c] + shift0[c];
    s0[i] = 1.f / (1.f + __expf(-g));
  } else {
    int j = i - half_n;
    int k = (j >> 8) & 127;
    int b = j >> 15;                            // 128*256 per batch
    float g = G1[j] * scale1[k] + shift1[k];
    float sv = 1.f / (1.f + __expf(-g));
    s1[j] = sv;
    t1m[j] = kc[j] * sv * (float)kmask[b * 128 + k];
  }
}

// ---------------------------------------------------------------------------
// Kernel 4: u[b,d] = sum_k t1m[b,k,d]
// ---------------------------------------------------------------------------
__global__ void reduce_u(const float* __restrict__ t1m, float* __restrict__ u) {
  int b = blockIdx.x;
  int d = threadIdx.x;
  float s = 0.f;
  for (int k = 0; k < Ksz; ++k)
    s += t1m[(size_t)(b * Ksz + k) * Dsz + d];
  u[b * Dsz + d] = s;
}

// ---------------------------------------------------------------------------
// Kernel 5: output_res = tanh(cmask * s0 * u / klen)
// ---------------------------------------------------------------------------
__global__ void out0_kernel(const float* __restrict__ s0, const float* __restrict__ u,
                            const int* __restrict__ cmask, const int* __restrict__ klen,
                            float* __restrict__ out0) {
  int i = blockIdx.x * 256 + threadIdx.x;       // 524288
  int b = i >> 15;
  int c = (i >> 8) & 127;
  int d = i & 255;
  float cm = (float)cmask[b * Csz + c];
  float v = cm * s0[i] * u[b * Dsz + d] / (float)klen[b];
  out0[i] = tanhf(v);
}

// ---------------------------------------------------------------------------
// Kernel 6: attention_res_masked[b,c,k,d] = t1m[b,k,d] * (s0*cmask)[b,c,d]
// 268 MB of write-once stores (> 192 MB L2): issue with non-temporal hint so
// the hot t1m/s0 working set (4 MB, reused 128x) stays L2-resident.
// ---------------------------------------------------------------------------
__global__ void bigout(const float* __restrict__ s0, const float* __restrict__ t1m,
                       const int* __restrict__ cmask, float* __restrict__ out1) {
  int bc = blockIdx.x;                  // b*128 + c
  int b = bc >> 7;
  int t = threadIdx.x;
  int d4 = t & 63;                      // float4 index along D
  int ks = t >> 6;                      // k sub-lane 0..3
  float cm = (float)cmask[bc];

  v4f sv = ((const v4f*)(s0 + (size_t)bc * Dsz))[d4];
  sv *= cm;

  v4f* outp = (v4f*)(out1 + (size_t)bc * Ksz * Dsz);
  const v4f* tp = (const v4f*)(t1m + (size_t)b * Ksz * Dsz);
  for (int k = ks; k < Ksz; k += 4) {
    v4f tv = tp[k * (Dsz / 4) + d4];
    v4f o = tv * sv;
    __builtin_nontemporal_store(o, &outp[k * (Dsz / 4) + d4]);
  }
}

// ---------------------------------------------------------------------------
// Kernel 7: attention_weight.mean(-1)[b,c,k] = (1/D) * dot(s0[b,c,:], s1[b,k,:])
// Batched GEMM via V_WMMA_F32_16X16X4_F32 (1024 tiles, one per wave).
// ---------------------------------------------------------------------------
__global__ void awmean_wmma(const float* __restrict__ s0, const float* __restrict__ s1,
                            float* __restrict__ out2) {
  int wave = threadIdx.x >> 5;
  int lane = threadIdx.x & 31;
  int tile = blockIdx.x * 4 + wave;     // b*64 + (c_tile*8 + k_tile)
  int b = tile >> 6;
  int t = tile & 63;
  int c_tile = t >> 3;
  int k_tile = t & 7;
  int half = lane >> 4;
  int l16  = lane & 15;

  const float* arow = s0 + (size_t)(b * Csz + c_tile * 16 + l16) * Dsz + 2 * half;
  const float* brow = s1 + (size_t)(b * Ksz + k_tile * 16 + l16) * Dsz + 2 * half;

  v8f acc = {};
  for (int d0 = 0; d0 < Dsz; d0 += 4) {
    v2f a = *(const v2f*)(arow + d0);
    v2f bb = *(const v2f*)(brow + d0);
    acc = __builtin_amdgcn_wmma_f32_16x16x4_f32(false, a, false, bb,
                                                (short)0, acc, false, false);
  }
  int col = k_tile * 16 + l16;
  int row0 = c_tile * 16 + 8 * half;
#pragma unroll
  for (int r = 0; r < 8; ++r)
    out2[(size_t)(b * Csz + row0 + r) * Ksz + col] = acc[r] * (1.f / 256.f);
}

extern "C" void kernel_launch(void* const* d_in, const int* in_sizes, int n_in,
                              void* d_out, int out_size, void* d_ws, size_t ws_size,
                              hipStream_t stream) {
  const float* q     = (const float*)d_in[0];   // (16,128,256)
  const float* kc    = (const float*)d_in[1];   // (16,128,256)
  const float* W0    = (const float*)d_in[2];   // (256,256)
  const float* W1    = (const float*)d_in[3];   // (256,256)
  const float* g0g   = (const float*)d_in[4];
  const float* g0b   = (const float*)d_in[5];
  const float* g1g   = (const float*)d_in[6];
  const float* g1b   = (const float*)d_in[7];
  const int*   cmask = (const int*)d_in[8];     // (16,128)
  const int*   kmask = (const int*)d_in[9];     // (16,128)
  const int*   klen  = (const int*)d_in[10];    // (16,)

  const size_t n_bcd = (size_t)Bsz * Csz * Dsz; // 524288

  float* ws = (float*)d_ws;
  float* G0     = ws;                 // 524288
  float* G1     = G0 + n_bcd;         // 524288
  float* s0     = G1 + n_bcd;         // 524288
  float* s1     = s0 + n_bcd;         // 524288
  float* t1m    = s1 + n_bcd;         // 524288
  float* u      = t1m + n_bcd;        // 4096
  float* scale0 = u + (size_t)Bsz * Dsz;
  float* shift0 = scale0 + 128;
  float* scale1 = shift0 + 128;
  float* shift1 = scale1 + 128;

  float* out0 = (float*)d_out;                      // (16,128,256)
  float* out1 = out0 + n_bcd;                       // (16,128,128,256)
  float* out2 = out1 + (size_t)Bsz * Csz * Ksz * Dsz; // (16,128,128)

  // GEMMs: 2048 tiles each, 4 waves per 128-thread block.
  gemm_f32_wmma<<<512, 128, 0, stream>>>(q,  W0, G0);
  gemm_f32_wmma<<<512, 128, 0, stream>>>(kc, W1, G1);

  bn_stats<<<256, 256, 0, stream>>>(G0, G1, g0g, g0b, g1g, g1b,
                                    scale0, shift0, scale1, shift1);

  bn_sig_apply<<<4096, 256, 0, stream>>>(G0, G1, kc, scale0, shift0,
                                         scale1, shift1, kmask, s0, s1, t1m);

  reduce_u<<<Bsz, Dsz, 0, stream>>>(t1m, u);

  out0_kernel<<<2048, 256, 0, stream>>>(s0, u, cmask, klen, out0);

  bigout<<<Bsz * Csz, 256, 0, stream>>>(s0, t1m, cmask, out1);

  awmean_wmma<<<256, 128, 0, stream>>>(s0, s1, out2);
}